// YoloLoss_29154238005278
// MI455X (gfx1250) — compile-verified
//
#include <hip/hip_runtime.h>
#include <hip/hip_bf16.h>
#include <stdint.h>

// YOLO loss, MI455X (gfx1250). Memory-bound streaming reduction:
//   192.7 MB @ 23.3 TB/s  =>  ~8.3 us floor.
// CDNA5 paths used:
//   - global_load_async_to_lds_b128 (ASYNCcnt) tile staging: coalesced 16B/lane
//     copies instead of 120B/cell strided lane reads.
//   - V_WMMA_F32_16X16X4_F32 ones-matmul wave reduction for the block partial
//     (matrix-engine reduction: D[m,n] = v[m] + v[m+16] with B = ones).

#define CELL_F   30                 // floats per cell
#define TILE     256                // cells per block (cells % 256 == 0 for this shape)
#define TILE_F   (TILE * CELL_F)    // 7680 floats per tensor tile
#define THREADS  256
#define SDIM     14

typedef float v2f __attribute__((ext_vector_type(2)));
typedef float v8f __attribute__((ext_vector_type(8)));

__device__ __forceinline__ void async_load_b128(unsigned lds_off, unsigned long long gaddr) {
  // gfx1250: copies 16 bytes from global memory to LDS, tracked by ASYNCcnt
  asm volatile("global_load_async_to_lds_b128 %0, %1, off"
               :: "v"(lds_off), "v"(gaddr) : "memory");
}

__device__ __forceinline__ void wait_async0() {
#if __has_builtin(__builtin_amdgcn_s_wait_asynccnt)
  __builtin_amdgcn_s_wait_asynccnt(0);
#else
  asm volatile("s_wait_asynccnt 0" ::: "memory");
#endif
}

__global__ __launch_bounds__(THREADS)
void yolo_main(const float* __restrict__ pred, const float* __restrict__ tgt,
               float* __restrict__ partials, int cells) {
  __shared__ float sp[TILE_F];      // pred tile
  __shared__ float st[TILE_F];      // target tile
  __shared__ float swred[16];       // 8 waves x 2 half-wave sums

  const int tid       = threadIdx.x;
  const int tileStart = blockIdx.x * TILE;
  const int tileCells = (cells - tileStart < TILE) ? (cells - tileStart) : TILE;
  const int nbytes    = tileCells * (CELL_F * 4);   // full tiles: 30720 (16B multiple)

  // --- stage tile into LDS with async coalesced B128 copies ---
  const unsigned lp = (unsigned)(uintptr_t)&sp[0];  // low 32 bits of generic LDS ptr = LDS offset
  const unsigned lt = (unsigned)(uintptr_t)&st[0];
  const unsigned long long gp = (unsigned long long)(uintptr_t)(pred + (size_t)tileStart * CELL_F);
  const unsigned long long gt = (unsigned long long)(uintptr_t)(tgt  + (size_t)tileStart * CELL_F);

  for (int b = tid * 16; b < nbytes; b += THREADS * 16) {
    async_load_b128(lp + (unsigned)b, gp + (unsigned long long)b);
    async_load_b128(lt + (unsigned)b, gt + (unsigned long long)b);
  }
  wait_async0();
  __syncthreads();

  // --- per-cell loss (1 thread = 1 cell); LDS reads are bank-conflict-free ---
  float loss = 0.0f;
  if (tid < tileCells) {
    const float* p = &sp[tid * CELL_F];
    const float* t = &st[tid * CELL_F];
    const float invS = 1.0f / (float)SDIM;

    const float t4  = t[4];
    const float coo = (t4 > 0.0f) ? 1.0f : 0.0f;
    const float noo = 1.0f - coo;

    const float d4 = p[4] - t4;
    const float d9 = p[9] - t[9];
    const float noo_l = d4 * d4 + d9 * d9;

    float cls = 0.0f;
#pragma unroll
    for (int k = 10; k < 30; ++k) { const float d = p[k] - t[k]; cls += d * d; }

    // target box 0 corners (xyxy)
    const float tbx0 = t[0] * invS - 0.5f * t[2];
    const float tbx1 = t[0] * invS + 0.5f * t[2];
    const float tby0 = t[1] * invS - 0.5f * t[3];
    const float tby1 = t[1] * invS + 0.5f * t[3];
    const float tA   = (tbx1 - tbx0) * (tby1 - tby0);

    float iou[2];
#pragma unroll
    for (int b = 0; b < 2; ++b) {
      const float* pb = p + 5 * b;
      const float ax0 = pb[0] * invS - 0.5f * pb[2];
      const float ax1 = pb[0] * invS + 0.5f * pb[2];
      const float ay0 = pb[1] * invS - 0.5f * pb[3];
      const float ay1 = pb[1] * invS + 0.5f * pb[3];
      const float ltx = fmaxf(ax0, tbx0), lty = fmaxf(ay0, tby0);
      const float rbx = fminf(ax1, tbx1), rby = fminf(ay1, tby1);
      const float w = fmaxf(rbx - ltx, 0.0f), h = fmaxf(rby - lty, 0.0f);
      const float inter = w * h;
      const float a1 = (ax1 - ax0) * (ay1 - ay0);
      iou[b] = inter / (a1 + tA - inter);
    }

    const int   idx     = (iou[1] > iou[0]) ? 1 : 0;   // jnp.argmax tie -> 0
    const float max_iou = fmaxf(iou[0], iou[1]);
    const float* rp = p + 5 * idx;
    const float* rt = t + 5 * idx;
    const float nr_conf = p[5 * (1 - idx) + 4];

    const float dc = rp[4] - max_iou;
    const float contain = dc * dc;

    const float dx = rp[0] - rt[0];
    const float dy = rp[1] - rt[1];
    const float dw = sqrtf(rp[2]) - sqrtf(rt[2]);
    const float dh = sqrtf(rp[3]) - sqrtf(rt[3]);
    const float loc = dx * dx + dy * dy + dw * dw + dh * dh;

    loss = coo * (5.0f * loc + 2.0f * contain + nr_conf * nr_conf + cls)
         + 0.5f * noo * noo_l;
  }

  // --- wave reduction on the matrix engine: D = A(16x4) * ones(4x16) ---
  // A layout (32-bit, 16x4): VGPR0 lanes0-15 -> K=0, lanes16-31 -> K=2; VGPR1 -> K=1/K=3.
  // With a.y = 0:  D[m,n] = v[m] + v[m+16]  (replicated over n).
  // D layout: VGPR r holds M=r (lanes 0-15) and M=8+r (lanes 16-31).
  // EXEC is all-ones here (uniform control flow across the block).
  v2f a; a.x = loss; a.y = 0.0f;
  v2f b1; b1.x = 1.0f; b1.y = 1.0f;
  v8f c = {};
  v8f d = __builtin_amdgcn_wmma_f32_16x16x4_f32(
      /*neg_a=*/false, a, /*neg_b=*/false, b1,
      /*c_mod=*/(short)0, c, /*reuse_a=*/false, /*reuse_b=*/false);
  const float s8 = ((d[0] + d[1]) + (d[2] + d[3])) + ((d[4] + d[5]) + (d[6] + d[7]));
  // lane 0 holds rows 0-7 (= lanes 0..7, 16..23 of v), lane 16 holds rows 8-15 (= lanes 8..15, 24..31).
  const int lane = tid & 31;
  const int wave = tid >> 5;
  if ((lane & 15) == 0) swred[wave * 2 + (lane >> 4)] = s8;
  __syncthreads();
  if (tid == 0) {
    float tsum = 0.0f;
#pragma unroll
    for (int i = 0; i < 16; ++i) tsum += swred[i];
    partials[blockIdx.x] = tsum;
  }
}

__global__ __launch_bounds__(256)
void yolo_finish(const float* __restrict__ partials, int nblocks,
                 float* __restrict__ out, float invN) {
  __shared__ float sred[256];
  float s = 0.0f;
  for (int i = threadIdx.x; i < nblocks; i += 256) s += partials[i];
  sred[threadIdx.x] = s;
  __syncthreads();
#pragma unroll
  for (int k = 128; k > 0; k >>= 1) {
    if (threadIdx.x < k) sred[threadIdx.x] += sred[threadIdx.x + k];
    __syncthreads();
  }
  if (threadIdx.x == 0) out[0] = sred[0] * invN;
}

extern "C" void kernel_launch(void* const* d_in, const int* in_sizes, int n_in,
                              void* d_out, int out_size, void* d_ws, size_t ws_size,
                              hipStream_t stream) {
  const float* pred = (const float*)d_in[0];
  const float* tgt  = (const float*)d_in[1];

  const int cells   = in_sizes[0] / CELL_F;              // 802816 (divisible by 256)
  const int nblocks = (cells + TILE - 1) / TILE;         // 3136
  const int N       = cells / (SDIM * SDIM);             // 4096

  float* partials = (float*)d_ws;                        // 3136 floats of scratch

  yolo_main<<<nblocks, THREADS, 0, stream>>>(pred, tgt, partials, cells);
  yolo_finish<<<1, 256, 0, stream>>>(partials, nblocks, (float*)d_out, 1.0f / (float)N);
}